// SLiCE_37761352466575
// MI455X (gfx1250) — compile-verified
//
#include <hip/hip_runtime.h>
#include <hip/hip_bf16.h>

// ---------------------------------------------------------------- constants
#define BATCH   32
#define SEQT    4096
#define DIN     128
#define KIN     130            // D_IN + 2
#define KPAD    132            // padded K (multiple of 4)
#define HID     256
#define NBLK    64             // G: number of 4x4 diagonal blocks
#define NOUT    1280           // HIDDEN*BLOCK + HIDDEN = 1024 + 256
#define CHUNK   64             // scan chunk length
#define NCH     (SEQT / CHUNK) // 64 chunks
#define NTASK   (BATCH * NCH * NBLK) // 131072 chunk-tasks
#define NROWS   (BATCH * SEQT)       // 131072 GEMM rows

static constexpr float kScale = 1.0f / 40.0f;

typedef float v2f  __attribute__((ext_vector_type(2)));
typedef float v4f  __attribute__((ext_vector_type(4)));
typedef float v8f  __attribute__((ext_vector_type(8)));

// ---------------------------------------------------------------- kernel 1
// Build padded, transposed, PAIR-INTERLEAVED, PRE-SCALED weight matrix:
//   Wtp[(p*NOUT + n)*2 + r] = SCALE * W[k=2p+r][n]  for k<130, else 0
//   where W[k][n] = n<1024 ? W_A[n][k] : W_B[n-1024][k]
// SCALE is folded here ((inp*S)@W^T == inp@(S*W)^T), so the GEMM's A-fragments
// are raw loads feeding WMMA directly. A lane's B-fragment is ONE float2 load.
__global__ __launch_bounds__(256) void prep_wt_kernel(const float* __restrict__ WA,
                                                      const float* __restrict__ WB,
                                                      float* __restrict__ Wtp) {
    int idx = blockIdx.x * blockDim.x + threadIdx.x;
    if (idx >= KPAD * NOUT) return;
    int k = idx / NOUT;
    int n = idx - k * NOUT;
    float v = 0.0f;
    if (k < KIN) {
        v = (n < 1024) ? WA[(size_t)n * KIN + k] : WB[(size_t)(n - 1024) * KIN + k];
        v *= kScale;
    }
    Wtp[((size_t)(k >> 1) * NOUT + n) * 2 + (k & 1)] = v;
}

// ---------------------------------------------------------------- kernel 2
// WMMA GEMM: coef[m][n] = sum_k inp[m][k] * (S*W)[k][n],  m in [0,131072)
// inp[m][k]: k==0 -> 1 ; k==1 -> t+1 ; else X[m][k-2]   (scale lives in weights)
// One wave per 16x64 output tile (4 f32 accumulators), 8 waves / block.
// k=0 and k=128 iterations are peeled so the steady-state loop is branch-free.
__global__ __launch_bounds__(256) void gemm_wmma_kernel(const float* __restrict__ X,
                                                        const float* __restrict__ Wtp,
                                                        float* __restrict__ coef) {
    const int lane  = threadIdx.x & 31;
    const int wave  = threadIdx.x >> 5;
    const int l15   = lane & 15;
    const bool hi   = lane >= 16;           // lanes 16..31 handle K slots +2,+3
    const int mtile = blockIdx.x * 8 + wave;
    const int m0    = mtile * 16;
    const int n0    = blockIdx.y * 64;

    const int m = m0 + l15;                 // row this lane feeds for the A-frag
    const int t = m & (SEQT - 1);           // time index within batch
    const float* __restrict__ xrow = X + (size_t)m * DIN;
    // steady-state A-frag base: lo lanes read X cols k-2,k-1 ; hi lanes k,k+1
    const float* __restrict__ xg = xrow + (hi ? 0 : -2);
    // B-frag base in pair-interleaved weights: pair p = k/2 + (hi?1:0), column n0+l15
    const size_t wlane = ((size_t)(hi ? 1 : 0) * NOUT + n0 + l15) * 2;

    v8f acc0 = {}; v8f acc1 = {}; v8f acc2 = {}; v8f acc3 = {};

    // ---- helper macro: one K-step given A-fragment `af` and pair index base kp=k/2
#define GEMM_STEP(af, kp)                                                          \
    {                                                                              \
        const float* __restrict__ wb = Wtp + (size_t)(kp) * (NOUT * 2) + wlane;    \
        const v2f b0 = *(const v2f*)(wb);                                          \
        const v2f b1 = *(const v2f*)(wb + 32);                                     \
        const v2f b2 = *(const v2f*)(wb + 64);                                     \
        const v2f b3 = *(const v2f*)(wb + 96);                                     \
        acc0 = __builtin_amdgcn_wmma_f32_16x16x4_f32(false, af, false, b0, (short)0, acc0, false, false); \
        acc1 = __builtin_amdgcn_wmma_f32_16x16x4_f32(false, af, false, b1, (short)0, acc1, false, false); \
        acc2 = __builtin_amdgcn_wmma_f32_16x16x4_f32(false, af, false, b2, (short)0, acc2, false, false); \
        acc3 = __builtin_amdgcn_wmma_f32_16x16x4_f32(false, af, false, b3, (short)0, acc3, false, false); \
    }

    // ---- peeled k = 0: lo lanes -> (1, t+1), hi lanes -> (X[0], X[1])
    {
        const v2f x01 = *(const v2f*)(xrow);      // always in-bounds, all lanes
        v2f af;
        af.x = hi ? x01.x : 1.0f;
        af.y = hi ? x01.y : (float)(t + 1);
        GEMM_STEP(af, 0)
    }

    // ---- steady state k = 4 .. 124 (31 branch-free iterations)
    for (int k = 4; k <= 124; k += 4) {
        const v2f af = *(const v2f*)(xg + k);     // raw X feeds WMMA directly
        GEMM_STEP(af, (k >> 1))
    }

    // ---- peeled k = 128: lo lanes -> (X[126], X[127]), hi lanes -> padded zeros
    {
        const v2f xl = *(const v2f*)(xrow + 126); // always in-bounds, all lanes
        v2f af;
        af.x = hi ? 0.0f : xl.x;
        af.y = hi ? 0.0f : xl.y;
        GEMM_STEP(af, 64)
    }
#undef GEMM_STEP

    // ---- store C/D tile: VGPR r, lanes 0-15 -> M=r, lanes 16-31 -> M=r+8, N=lane&15
    const int mbase = m0 + (hi ? 8 : 0);
#pragma unroll
    for (int r = 0; r < 8; ++r) {
        const size_t off = (size_t)(mbase + r) * NOUT + n0 + l15;
        coef[off]      = acc0[r];
        coef[off + 16] = acc1[r];
        coef[off + 32] = acc2[r];
        coef[off + 48] = acc3[r];
    }
}

// ---------------------------------------------------------------- kernel 3
// Per (b, chunk, block) compose the chunk's affine map: y -> P*y + q with
// P = prod_{t in chunk} (I + A_t), accumulated left-to-right.
__global__ __launch_bounds__(256) void chunk_compose_kernel(const float* __restrict__ coef,
                                                            float* __restrict__ summ) {
    const int task = blockIdx.x * blockDim.x + threadIdx.x;
    const int g = task & (NBLK - 1);
    const int c = (task >> 6) & (NCH - 1);
    const int b = task >> 12;

    float P[4][4] = {{1,0,0,0},{0,1,0,0},{0,0,1,0},{0,0,0,1}};
    float q[4]    = {0,0,0,0};

    const float* __restrict__ rowbase = coef + (size_t)(b * SEQT + c * CHUNK) * NOUT;

    for (int tt = 0; tt < CHUNK; ++tt) {
        const float* __restrict__ row = rowbase + (size_t)tt * NOUT;
        const v4f* __restrict__ ar = (const v4f*)(row + g * 16);
        const v4f A0 = ar[0], A1 = ar[1], A2 = ar[2], A3 = ar[3];
        const v4f Bv = *(const v4f*)(row + 1024 + g * 4);

        float np[4][4], nq[4];
#pragma unroll
        for (int j = 0; j < 4; ++j) {
            np[0][j] = P[0][j] + A0.x*P[0][j] + A0.y*P[1][j] + A0.z*P[2][j] + A0.w*P[3][j];
            np[1][j] = P[1][j] + A1.x*P[0][j] + A1.y*P[1][j] + A1.z*P[2][j] + A1.w*P[3][j];
            np[2][j] = P[2][j] + A2.x*P[0][j] + A2.y*P[1][j] + A2.z*P[2][j] + A2.w*P[3][j];
            np[3][j] = P[3][j] + A3.x*P[0][j] + A3.y*P[1][j] + A3.z*P[2][j] + A3.w*P[3][j];
        }
        nq[0] = q[0] + A0.x*q[0] + A0.y*q[1] + A0.z*q[2] + A0.w*q[3] + Bv.x;
        nq[1] = q[1] + A1.x*q[0] + A1.y*q[1] + A1.z*q[2] + A1.w*q[3] + Bv.y;
        nq[2] = q[2] + A2.x*q[0] + A2.y*q[1] + A2.z*q[2] + A2.w*q[3] + Bv.z;
        nq[3] = q[3] + A3.x*q[0] + A3.y*q[1] + A3.z*q[2] + A3.w*q[3] + Bv.w;
#pragma unroll
        for (int i = 0; i < 4; ++i) {
#pragma unroll
            for (int j = 0; j < 4; ++j) P[i][j] = np[i][j];
            q[i] = nq[i];
        }
    }
    // SoA store: 20 planes of NTASK floats (16 for P row-major, 4 for q)
#pragma unroll
    for (int i = 0; i < 4; ++i)
#pragma unroll
        for (int j = 0; j < 4; ++j)
            summ[(size_t)(i * 4 + j) * NTASK + task] = P[i][j];
#pragma unroll
    for (int i = 0; i < 4; ++i)
        summ[(size_t)(16 + i) * NTASK + task] = q[i];
}

// ---------------------------------------------------------------- kernel 4
// Sequential scan over chunk summaries per (b, block): emit chunk-start states.
__global__ __launch_bounds__(256) void summary_scan_kernel(const float* __restrict__ summ,
                                                           const float* __restrict__ init,
                                                           float* __restrict__ ystart) {
    const int task = blockIdx.x * blockDim.x + threadIdx.x; // BATCH*NBLK = 2048
    const int g = task & (NBLK - 1);
    const int b = task >> 6;

    float y[4];
#pragma unroll
    for (int i = 0; i < 4; ++i) y[i] = init[g * 4 + i];

    for (int c = 0; c < NCH; ++c) {
        const size_t sid = (size_t)((b * NCH + c) * NBLK) + g;
#pragma unroll
        for (int i = 0; i < 4; ++i) ystart[(size_t)i * NTASK + sid] = y[i];

        float P[4][4], q[4], ny[4];
#pragma unroll
        for (int i = 0; i < 4; ++i) {
#pragma unroll
            for (int j = 0; j < 4; ++j) P[i][j] = summ[(size_t)(i * 4 + j) * NTASK + sid];
            q[i] = summ[(size_t)(16 + i) * NTASK + sid];
        }
#pragma unroll
        for (int i = 0; i < 4; ++i)
            ny[i] = P[i][0]*y[0] + P[i][1]*y[1] + P[i][2]*y[2] + P[i][3]*y[3] + q[i];
#pragma unroll
        for (int i = 0; i < 4; ++i) y[i] = ny[i];
    }
}

// ---------------------------------------------------------------- kernel 5
// Replay each chunk from its start state and write outputs.
__global__ __launch_bounds__(256) void chunk_replay_kernel(const float* __restrict__ coef,
                                                           const float* __restrict__ ystart,
                                                           float* __restrict__ out) {
    const int task = blockIdx.x * blockDim.x + threadIdx.x;
    const int g = task & (NBLK - 1);
    const int c = (task >> 6) & (NCH - 1);
    const int b = task >> 12;

    float y0 = ystart[(size_t)0 * NTASK + task];
    float y1 = ystart[(size_t)1 * NTASK + task];
    float y2 = ystart[(size_t)2 * NTASK + task];
    float y3 = ystart[(size_t)3 * NTASK + task];

    const float* __restrict__ rowbase = coef + (size_t)(b * SEQT + c * CHUNK) * NOUT;
    float* __restrict__ outbase = out + (size_t)(b * SEQT + c * CHUNK) * HID + g * 4;

    for (int tt = 0; tt < CHUNK; ++tt) {
        const float* __restrict__ row = rowbase + (size_t)tt * NOUT;
        const v4f* __restrict__ ar = (const v4f*)(row + g * 16);
        const v4f A0 = ar[0], A1 = ar[1], A2 = ar[2], A3 = ar[3];
        const v4f Bv = *(const v4f*)(row + 1024 + g * 4);

        const float s0 = A0.x*y0 + A0.y*y1 + A0.z*y2 + A0.w*y3 + Bv.x;
        const float s1 = A1.x*y0 + A1.y*y1 + A1.z*y2 + A1.w*y3 + Bv.y;
        const float s2 = A2.x*y0 + A2.y*y1 + A2.z*y2 + A2.w*y3 + Bv.z;
        const float s3 = A3.x*y0 + A3.y*y1 + A3.z*y2 + A3.w*y3 + Bv.w;
        y0 += s0; y1 += s1; y2 += s2; y3 += s3;

        v4f o; o.x = y0; o.y = y1; o.z = y2; o.w = y3;
        *(v4f*)(outbase + (size_t)tt * HID) = o;
    }
}

// ---------------------------------------------------------------- launcher
extern "C" void kernel_launch(void* const* d_in, const int* in_sizes, int n_in,
                              void* d_out, int out_size, void* d_ws, size_t ws_size,
                              hipStream_t stream) {
    const float* X    = (const float*)d_in[0]; // (32, 4096, 128)
    const float* W_A  = (const float*)d_in[1]; // (1024, 130)
    const float* W_B  = (const float*)d_in[2]; // (256, 130)
    const float* init = (const float*)d_in[3]; // (256,)
    float* out = (float*)d_out;                // (32, 4096, 256)

    // workspace layout (floats)
    float* coef   = (float*)d_ws;                       // NROWS * NOUT   = 167,772,160
    float* Wtp    = coef + (size_t)NROWS * NOUT;        // KPAD * NOUT    = 168,960
    float* summ   = Wtp + (size_t)KPAD * NOUT;          // 20 * NTASK     = 2,621,440
    float* ystart = summ + (size_t)20 * NTASK;          // 4 * NTASK      = 524,288

    // 1) weight pad/transpose/pair-interleave/pre-scale
    prep_wt_kernel<<<(KPAD * NOUT + 255) / 256, 256, 0, stream>>>(W_A, W_B, Wtp);

    // 2) WMMA GEMM: grid = (M-tiles / 8 waves, N-groups of 64)
    gemm_wmma_kernel<<<dim3(NROWS / 16 / 8, NOUT / 64), 256, 0, stream>>>(X, Wtp, coef);

    // 3) chunk composition (131072 tasks)
    chunk_compose_kernel<<<NTASK / 256, 256, 0, stream>>>(coef, summ);

    // 4) summary scan (2048 tasks)
    summary_scan_kernel<<<(BATCH * NBLK) / 256, 256, 0, stream>>>(summ, init, ystart);

    // 5) chunk replay + output (131072 tasks)
    chunk_replay_kernel<<<NTASK / 256, 256, 0, stream>>>(coef, ystart, out);
}